// CostVolume2D_4947802325356
// MI455X (gfx1250) — compile-verified
//
#include <hip/hip_runtime.h>

typedef __attribute__((ext_vector_type(2))) float v2f;
typedef __attribute__((ext_vector_type(8))) float v8f;

#define BATCH   4
#define CHANS   64
#define HEIGHT  256
#define WIDTH   512
#define MAXD    48
#define NDISP   97              // 2*MAXD+1
#define HW      (HEIGHT*WIDTH)  // 131072
#define CHW     (CHANS*HW)      // 8388608
#define LDS_PITCH 17            // 16 + 1 pad -> conflict-free stride-16 scatter
#define WAVES_PER_BLOCK 8
#define TILES_PER_WAVE  4       // 32 M-tiles of 16 / 8 waves

__global__ __launch_bounds__(256) void cost_volume_wmma(
    const float* __restrict__ L, const float* __restrict__ R,
    float* __restrict__ out)
{
    __shared__ float lds[WAVES_PER_BLOCK * NDISP * LDS_PITCH]; // 52,768 B

    const int b    = blockIdx.x >> 8;    // B*H = 1024 blocks
    const int h    = blockIdx.x & 255;
    const int tid  = threadIdx.x;
    const int wave = tid >> 5;
    const int lane = tid & 31;
    const int lo   = lane & 15;          // M / N / column-in-tile index
    const int hi   = lane >> 4;          // lane-group: selects K pair / M half

    const long base0 = (long)b * CHW + (long)h * WIDTH;
    float* wlds = &lds[wave * (NDISP * LDS_PITCH)];

    for (int t = 0; t < TILES_PER_WAVE; ++t) {
        const int w0 = (wave * TILES_PER_WAVE + t) * 16;

        // ---- Stage all 16 A fragments (16x4 f32 tiles, K along channels) ----
        // lane holds A[m=lo, k=2*hi+r], r=0,1  ->  c = 4*ks + 2*hi + r
        v2f a[16];
        {
            const float* pa = L + base0 + (long)(2 * hi) * HW + (w0 + lo);
            #pragma unroll
            for (int ks = 0; ks < 16; ++ks) {
                a[ks].x = pa[0];
                a[ks].y = pa[HW];
                pa += 4 * (long)HW;
            }
        }

        // ---- 7 banded N-tiles: offsets -48..+48 ----
        #pragma unroll
        for (int o = 0; o < 7; ++o) {
            const int off = (o - 3) * 16;
            const int wq  = w0 + off + lo;               // w' column
            const bool inb = (wq >= 0) && (wq < WIDTH);
            const int wqc  = inb ? wq : 0;               // clamped, always in-bounds

            // Stage all 16 B fragments into distinct registers so the WMMA
            // chain below has no load/cndmask WAR interleaved with it.
            v2f bf[16];
            {
                const float* pb = R + base0 + (long)(2 * hi) * HW + wqc;
                #pragma unroll
                for (int ks = 0; ks < 16; ++ks) {
                    float b0 = pb[0];
                    float b1 = pb[HW];
                    bf[ks].x = inb ? b0 : 0.0f;          // zero-pad columns
                    bf[ks].y = inb ? b1 : 0.0f;
                    pb += 4 * (long)HW;
                }
            }

            // Two interleaved accumulate chains -> dependent-WMMA distance 2.
            v8f acc0 = {};
            v8f acc1 = {};
            #pragma unroll
            for (int ks = 0; ks < 8; ++ks) {
                acc0 = __builtin_amdgcn_wmma_f32_16x16x4_f32(
                    false, a[2 * ks + 0], false, bf[2 * ks + 0],
                    (short)0, acc0, false, false);
                acc1 = __builtin_amdgcn_wmma_f32_16x16x4_f32(
                    false, a[2 * ks + 1], false, bf[2 * ks + 1],
                    (short)0, acc1, false, false);
            }

            // ---- Diagonal scatter into LDS: d = m - n - off ----
            // vgpr r -> m = r + 8*hi, n = lo (C/D layout)
            #pragma unroll
            for (int r = 0; r < 8; ++r) {
                const int m = r + 8 * hi;
                const int d = m - lo - off;
                if (d >= -MAXD && d <= MAXD) {
                    wlds[(d + MAXD) * LDS_PITCH + m] =
                        (acc0[r] + acc1[r]) * (1.0f / 64.0f);
                }
            }
        }

        // ---- Drain LDS slab (97 x 16) with contiguous 64B global stores ----
        // LDS ops from the same wave are in-order; no barrier needed.
        // Output is write-once -> non-temporal, keep L2 for reused inputs.
        float* obase = out + (long)b * NDISP * HW + (long)h * WIDTH + w0;
        for (int it = 0; it < 49; ++it) {
            const int e  = it * 32 + lane;   // 1552 valid elements
            const int dd = e >> 4;
            const int j  = e & 15;
            if (dd < NDISP) {
                __builtin_nontemporal_store(wlds[dd * LDS_PITCH + j],
                                            &obase[(long)dd * HW + j]);
            }
        }
    }
}

extern "C" void kernel_launch(void* const* d_in, const int* in_sizes, int n_in,
                              void* d_out, int out_size, void* d_ws, size_t ws_size,
                              hipStream_t stream)
{
    const float* l_fmap = (const float*)d_in[0];
    const float* r_fmap = (const float*)d_in[1];
    // d_in[2] is max_disp (= 48), baked into the kernel as MAXD.
    float* out = (float*)d_out;

    dim3 grid(BATCH * HEIGHT);   // 1024 blocks, one (b,h) row-pair each
    dim3 block(256);             // 8 wave32
    cost_volume_wmma<<<grid, block, 0, stream>>>(l_fmap, r_fmap, out);
}